// EncapsulatedNTM_77000173683049
// MI455X (gfx1250) — compile-verified
//
#include <hip/hip_runtime.h>
#include <math.h>
#include <stdint.h>

// Problem constants (from reference)
#define B_    256
#define N_    2048
#define M_    64
#define CTRL_ 512
#define NIN_  256
#define NOUT_ 256
#define DIN_  320    // NIN + M
#define G4_   2048   // 4*CTRL
#define KCAT_ 576    // CTRL + M
#define PRW_  70     // M+6
#define PWW_  198    // 3M+6

typedef float v2f __attribute__((ext_vector_type(2)));
typedef float v8f __attribute__((ext_vector_type(8)));

__device__ __forceinline__ float sigmoidf_(float x) { return 1.0f / (1.0f + __expf(-x)); }
__device__ __forceinline__ float softplusf_(float x) { return x > 20.0f ? x : log1pf(__expf(x)); }

// ---------------------------------------------------------------------------
// f32 WMMA 16x16 tile:  D += A[a0:a0+16, :K] * W[n0:n0+16, :K]^T
// Layout (CDNA5 V_WMMA_F32_16X16X4_F32):
//   A 16x4: lane&15 = M row, (lane>>4)*2 = K pair -> v2f
//   B 4x16: lane&15 = N col, (lane>>4)*2 = K pair -> v2f  (B[k][n] = W[n][k])
//   C/D:    lane&15 = N, VGPR v -> M = (lane>>4)*8 + v
// ---------------------------------------------------------------------------
__device__ __forceinline__ v8f wmma_tile_f32(const float* __restrict__ A, int lda, int a0,
                                             const float* __restrict__ Wt, int ldw, int n0,
                                             int K, v8f acc, int lane) {
  const int r  = lane & 15;
  const int kb = (lane >> 4) << 1;
  const float* ap = A  + (size_t)(a0 + r) * lda + kb;
  const float* wp = Wt + (size_t)(n0 + r) * ldw + kb;
  for (int k = 0; k < K; k += 4) {
    v2f av = *(const v2f*)(ap + k);
    v2f bv = *(const v2f*)(wp + k);
    acc = __builtin_amdgcn_wmma_f32_16x16x4_f32(false, av, false, bv,
                                                (short)0, acc, false, false);
  }
  return acc;
}

// ---------------------------------------------------------------------------
// 0) inp = concat([x, prev_read], -1)   (B x 320)
// ---------------------------------------------------------------------------
__global__ void k_concat(const float* __restrict__ x, const float* __restrict__ pr,
                         float* __restrict__ inp) {
  int j = blockIdx.x * 256 + threadIdx.x;
  if (j >= B_ * DIN_) return;
  int b = j / DIN_, k = j % DIN_;
  inp[j] = (k < NIN_) ? x[b * NIN_ + k] : pr[b * M_ + (k - NIN_)];
}

// ---------------------------------------------------------------------------
// 1) gates = inp @ W_ih^T + h0 @ W_hh^T + b_ih + b_hh   (B x 2048), WMMA f32
//    grid: 2048 tiles of 16x16, 4 waves/block -> 512 blocks, no divergence.
// ---------------------------------------------------------------------------
__global__ __launch_bounds__(128) void k_gates_wmma(
    const float* __restrict__ inp, const float* __restrict__ h0,
    const float* __restrict__ W_ih, const float* __restrict__ W_hh,
    const float* __restrict__ b_ih, const float* __restrict__ b_hh,
    float* __restrict__ gates) {
  const int lane = threadIdx.x & 31;
  const int tile = blockIdx.x * 4 + (threadIdx.x >> 5);
  const int b0 = (tile & 15) * 16;
  const int g0 = (tile >> 4) * 16;
  v8f acc = {};
  acc = wmma_tile_f32(inp, DIN_,  b0, W_ih, DIN_,  g0, DIN_,  acc, lane);
  acc = wmma_tile_f32(h0,  CTRL_, b0, W_hh, CTRL_, g0, CTRL_, acc, lane);
  const int n  = lane & 15;
  const int mb = (lane >> 4) * 8;
  const float bias = b_ih[g0 + n] + b_hh[g0 + n];
  for (int v = 0; v < 8; ++v)
    gates[(size_t)(b0 + mb + v) * G4_ + g0 + n] = acc[v] + bias;
}

// ---------------------------------------------------------------------------
// 2) LSTM pointwise -> h1, c1 (also h1 into hcat[:,0:512])
// ---------------------------------------------------------------------------
__global__ void k_lstm(const float* __restrict__ gates, const float* __restrict__ c0,
                       float* __restrict__ h1, float* __restrict__ c1,
                       float* __restrict__ hcat) {
  int j = blockIdx.x * 256 + threadIdx.x;
  if (j >= B_ * CTRL_) return;
  int b = j >> 9, c = j & (CTRL_ - 1);
  const float* gb = gates + (size_t)b * G4_;
  float gi = gb[c], gf = gb[CTRL_ + c], gg = gb[2 * CTRL_ + c], go = gb[3 * CTRL_ + c];
  float cv = sigmoidf_(gf) * c0[j] + sigmoidf_(gi) * tanhf(gg);
  float hv = sigmoidf_(go) * tanhf(cv);
  h1[j] = hv; c1[j] = cv;
  hcat[(size_t)b * KCAT_ + c] = hv;
}

// ---------------------------------------------------------------------------
// 3) pr = h1 @ W_rh^T + b_rh (B x 70); pw = h1 @ W_wh^T + b_wh (B x 198)
// ---------------------------------------------------------------------------
__global__ void k_heads(const float* __restrict__ hcat,
                        const float* __restrict__ W_rh, const float* __restrict__ b_rh,
                        const float* __restrict__ W_wh, const float* __restrict__ b_wh,
                        float* __restrict__ pr, float* __restrict__ pw) {
  int j = blockIdx.x * 256 + threadIdx.x;
  if (j >= B_ * (PRW_ + PWW_)) return;
  int b = j / (PRW_ + PWW_), tt = j % (PRW_ + PWW_);
  const float* h = hcat + (size_t)b * KCAT_;
  const float* wrow; float bias; float* dst;
  if (tt < PRW_) { wrow = W_rh + (size_t)tt * CTRL_; bias = b_rh[tt]; dst = pr + b * PRW_ + tt; }
  else { int u = tt - PRW_; wrow = W_wh + (size_t)u * CTRL_; bias = b_wh[u]; dst = pw + b * PWW_ + u; }
  float acc = 0.f;
  for (int k = 0; k < CTRL_; k += 4) {
    float4 hv = *(const float4*)(h + k);
    float4 wv = *(const float4*)(wrow + k);
    acc += hv.x * wv.x + hv.y * wv.y + hv.z * wv.z + hv.w * wv.w;
  }
  *dst = acc + bias;
}

// ---------------------------------------------------------------------------
// 4) NTM addressing, one block per batch row. Streams memory ONCE computing
//    both heads' sims + norms, then softmax/interp/shift/sharpen in LDS,
//    then an L2-warm second pass for r = w_r . memory.
// ---------------------------------------------------------------------------
__global__ __launch_bounds__(256) void k_address(
    const float* __restrict__ mem, const float* __restrict__ pr, const float* __restrict__ pw,
    const float* __restrict__ w_read, const float* __restrict__ w_write,
    float* __restrict__ wr_out, float* __restrict__ ww_out,
    float* __restrict__ r_out, float* __restrict__ hcat) {
  __shared__ float s_simr[N_];
  __shared__ float s_simw[N_];
  __shared__ float s_wbuf[N_];
  __shared__ float s_red[256];
  __shared__ float s_kr[M_], s_kw[M_];
  __shared__ float s_par[12];
  __shared__ float s_kn[2];
  __shared__ float s_rp[8][M_];
  const int b = blockIdx.x, t = threadIdx.x;
  const int w = t >> 5, l = t & 31;
  const float* prb = pr + (size_t)b * PRW_;
  const float* pwb = pw + (size_t)b * PWW_;
  if (t < M_) { s_kr[t] = prb[t] + 1e-16f; s_kw[t] = pwb[t] + 1e-16f; }
  if (t == 0) {
    s_par[0] = softplusf_(prb[M_]);
    s_par[1] = sigmoidf_(prb[M_ + 1]);
    float a0 = prb[M_ + 2], a1 = prb[M_ + 3], a2 = prb[M_ + 4];
    float mx = fmaxf(a0, fmaxf(a1, a2));
    float e0 = __expf(a0 - mx), e1 = __expf(a1 - mx), e2 = __expf(a2 - mx), es = e0 + e1 + e2;
    s_par[2] = e0 / es; s_par[3] = e1 / es; s_par[4] = e2 / es;
    s_par[5] = 1.0f + softplusf_(prb[M_ + 5]);
  } else if (t == 1) {
    s_par[6] = softplusf_(pwb[M_]);
    s_par[7] = sigmoidf_(pwb[M_ + 1]);
    float a0 = pwb[M_ + 2], a1 = pwb[M_ + 3], a2 = pwb[M_ + 4];
    float mx = fmaxf(a0, fmaxf(a1, a2));
    float e0 = __expf(a0 - mx), e1 = __expf(a1 - mx), e2 = __expf(a2 - mx), es = e0 + e1 + e2;
    s_par[8] = e0 / es; s_par[9] = e1 / es; s_par[10] = e2 / es;
    s_par[11] = 1.0f + softplusf_(pwb[M_ + 5]);
  }
  __syncthreads();
  if (t == 0) {
    float sr = 0.f, sw = 0.f;
    for (int i = 0; i < M_; ++i) { sr += s_kr[i] * s_kr[i]; sw += s_kw[i] * s_kw[i]; }
    s_kn[0] = sqrtf(sr); s_kn[1] = sqrtf(sw);
  }
  __syncthreads();
  // Phase 1: cosine sims for both heads in one streaming pass (wave per row).
  const float knr = s_kn[0], knw = s_kn[1];
  const float* mb = mem + (size_t)b * N_ * M_;
  for (int n = w; n < N_; n += 8) {
    v2f mm = *(const v2f*)(mb + (size_t)n * M_ + 2 * l);
    float mx = mm.x + 1e-16f, my = mm.y + 1e-16f;
    float dr = mx * s_kr[2 * l] + my * s_kr[2 * l + 1];
    float dw = mx * s_kw[2 * l] + my * s_kw[2 * l + 1];
    float nn = mx * mx + my * my;
    for (int off = 16; off >= 1; off >>= 1) {
      dr += __shfl_xor(dr, off, 32);
      dw += __shfl_xor(dw, off, 32);
      nn += __shfl_xor(nn, off, 32);
    }
    if (l == 0) {
      float nm = sqrtf(nn);
      s_simr[n] = dr / fmaxf(nm * knr, 1e-8f);
      s_simw[n] = dw / fmaxf(nm * knw, 1e-8f);
    }
  }
  __syncthreads();
  // Phase 2: softmax -> interpolate -> circular shift -> sharpen, per head.
  for (int h = 0; h < 2; ++h) {
    float* sim = h ? s_simw : s_simr;
    const float* wprev = (h ? w_write : w_read) + (size_t)b * N_;
    float* outp = (h ? ww_out : wr_out) + (size_t)b * N_;
    const float beta = s_par[h * 6 + 0], gg = s_par[h * 6 + 1];
    const float sh0 = s_par[h * 6 + 2], sh1 = s_par[h * 6 + 3], sh2 = s_par[h * 6 + 4];
    const float gamma = s_par[h * 6 + 5];
    float lm = -3.4e38f;
    for (int n = t; n < N_; n += 256) lm = fmaxf(lm, beta * sim[n]);
    s_red[t] = lm; __syncthreads();
    for (int s = 128; s; s >>= 1) { if (t < s) s_red[t] = fmaxf(s_red[t], s_red[t + s]); __syncthreads(); }
    const float mx = s_red[0]; __syncthreads();
    float ls = 0.f;
    for (int n = t; n < N_; n += 256) { float e = __expf(beta * sim[n] - mx); sim[n] = e; ls += e; }
    s_red[t] = ls; __syncthreads();
    for (int s = 128; s; s >>= 1) { if (t < s) s_red[t] += s_red[t + s]; __syncthreads(); }
    const float Z = s_red[0]; __syncthreads();
    for (int n = t; n < N_; n += 256) {
      float wc = sim[n] / Z;
      sim[n] = gg * wc + (1.0f - gg) * wprev[n];
    }
    __syncthreads();
    float lp = 0.f;
    for (int n = t; n < N_; n += 256) {
      float wh = sim[(n + N_ - 1) & (N_ - 1)] * sh0 + sim[n] * sh1 + sim[(n + 1) & (N_ - 1)] * sh2;
      float p = powf(wh, gamma);
      s_wbuf[n] = p; lp += p;
    }
    s_red[t] = lp; __syncthreads();
    for (int s = 128; s; s >>= 1) { if (t < s) s_red[t] += s_red[t + s]; __syncthreads(); }
    const float S = s_red[0] + 1e-16f; __syncthreads();
    for (int n = t; n < N_; n += 256) {
      float wv = s_wbuf[n] / S;
      outp[n] = wv;
      sim[n] = wv;                      // final w_r kept in s_simr for phase 3
    }
    __syncthreads();
  }
  // Phase 3: r = sum_n w_r[n] * memory[b,n,:]  (slice is warm in L2)
  float rx = 0.f, ry = 0.f;
  for (int n = w; n < N_; n += 8) {
    v2f mm = *(const v2f*)(mb + (size_t)n * M_ + 2 * l);
    float wv = s_simr[n];
    rx += wv * mm.x; ry += wv * mm.y;
  }
  s_rp[w][2 * l] = rx; s_rp[w][2 * l + 1] = ry;
  __syncthreads();
  if (t < M_) {
    float acc = 0.f;
    for (int i = 0; i < 8; ++i) acc += s_rp[i][t];
    r_out[(size_t)b * M_ + t] = acc;
    hcat[(size_t)b * KCAT_ + CTRL_ + t] = acc;
  }
}

// ---------------------------------------------------------------------------
// 5) out = sigmoid([h1|r] @ W_o^T + b_o)   (B x 256), WMMA f32
// ---------------------------------------------------------------------------
__global__ __launch_bounds__(128) void k_out_wmma(
    const float* __restrict__ hcat, const float* __restrict__ W_o,
    const float* __restrict__ b_o, float* __restrict__ out) {
  const int lane = threadIdx.x & 31;
  const int tile = blockIdx.x * 4 + (threadIdx.x >> 5);
  const int b0 = (tile & 15) * 16;
  const int o0 = (tile >> 4) * 16;
  v8f acc = {};
  acc = wmma_tile_f32(hcat, KCAT_, b0, W_o, KCAT_, o0, KCAT_, acc, lane);
  const int n  = lane & 15;
  const int mb = (lane >> 4) * 8;
  const float bias = b_o[o0 + n];
  for (int v = 0; v < 8; ++v)
    out[(size_t)(b0 + mb + v) * NOUT_ + o0 + n] = sigmoidf_(acc[v] + bias);
}

// ---------------------------------------------------------------------------
// 6) mem1 = mem*(1 - w_w e) + w_w a, stored TRANSPOSED as (B, M, N).
//    64x64 tile filled by GLOBAL_LOAD_ASYNC_TO_LDS_B32 (CDNA5 async path,
//    tracked by ASYNCcnt; no VGPR staging, no ds_store issue slots),
//    row stride 65 -> conflict-free transposed readout, fused erase/add,
//    coalesced stores of output_memory.
// ---------------------------------------------------------------------------
__global__ __launch_bounds__(256) void k_update_mem(
    const float* __restrict__ mem, const float* __restrict__ pw,
    const float* __restrict__ ww, float* __restrict__ omem) {
  __shared__ float s_t[64 * 65];
  __shared__ float s_ww[64], s_e[64], s_a[64];
  const int b  = blockIdx.x >> 5;
  const int n0 = (blockIdx.x & 31) * 64;
  const int t  = threadIdx.x;
  const float* mbp = mem + ((size_t)b * N_ + n0) * M_;
  // Async-copy the raw 64(n) x 64(m) tile into LDS. Per ISA the per-lane LDS
  // destination byte address comes from a VGPR; the low 32 bits of a generic
  // pointer to LDS are exactly that address (flat->LDS uses addr[31:0]).
  for (int it = 0; it < 16; ++it) {
    int idx = it * 256 + t;                 // idx = row*64 + col
    int row = idx >> 6, col = idx & 63;
    uint64_t gaddr = (uint64_t)(uintptr_t)(mbp + idx);
    uint32_t laddr = (uint32_t)(uintptr_t)(s_t + row * 65 + col);
    asm volatile("global_load_async_to_lds_b32 %0, %1, off"
                 :: "v"(laddr), "v"(gaddr)
                 : "memory");
  }
  if (t < 64) {
    s_ww[t] = ww[(size_t)b * N_ + n0 + t];
    s_e[t]  = sigmoidf_(pw[(size_t)b * PWW_ + PRW_ + t]);
    s_a[t]  = pw[(size_t)b * PWW_ + PRW_ + M_ + t];
  }
  asm volatile("s_wait_asynccnt 0x0" ::: "memory");
  __syncthreads();
  // Transposed readout + fused update: thread t -> column j = t&63 of the
  // tile (n index), m-range [(t>>6)*16, +16). LDS bank = (65j+m)%64 = (j+m)%64
  // -> 32 consecutive j per wave hit 32 distinct banks. Stores: 32 consecutive
  // n per wave per m -> 128B coalesced.
  float* ob = omem + (size_t)b * M_ * N_ + n0;
  const int j  = t & 63;
  const int m0 = (t >> 6) * 16;
  const float wwv = s_ww[j];
  for (int i = 0; i < 16; ++i) {
    int m = m0 + i;
    float v = s_t[j * 65 + m];
    ob[(size_t)m * N_ + j] = v * (1.0f - wwv * s_e[m]) + wwv * s_a[m];
  }
}

// ---------------------------------------------------------------------------
// 7) residual_feat = pick_last(ind==1) - pick_last(ind==-1)
// ---------------------------------------------------------------------------
__global__ __launch_bounds__(256) void k_residual(
    const int* __restrict__ ind, const float* __restrict__ omem, float* __restrict__ res) {
  __shared__ int s_ip[256], s_im[256];
  const int b = blockIdx.x, t = threadIdx.x;
  int ip = -1, im = -1;
  for (int n = t; n < N_; n += 256) {
    int v = ind[(size_t)b * N_ + n];
    if (v == 1) ip = n;       // ascending loop -> ends at this thread's max
    if (v == -1) im = n;
  }
  s_ip[t] = ip; s_im[t] = im;
  __syncthreads();
  for (int s = 128; s; s >>= 1) {
    if (t < s) { s_ip[t] = max(s_ip[t], s_ip[t + s]); s_im[t] = max(s_im[t], s_im[t + s]); }
    __syncthreads();
  }
  const int np = s_ip[0], nm = s_im[0];
  if (t < M_) {
    const float* ob = omem + (size_t)b * M_ * N_;
    float vp = (np >= 0) ? ob[(size_t)t * N_ + np] : 0.0f;
    float vm = (nm >= 0) ? ob[(size_t)t * N_ + nm] : 0.0f;
    res[(size_t)b * M_ + t] = vp - vm;
  }
}

// ---------------------------------------------------------------------------
extern "C" void kernel_launch(void* const* d_in, const int* in_sizes, int n_in,
                              void* d_out, int out_size, void* d_ws, size_t ws_size,
                              hipStream_t stream) {
  const float* x       = (const float*)d_in[0];
  const float* mem     = (const float*)d_in[1];
  const float* prev_r  = (const float*)d_in[2];
  const float* h0      = (const float*)d_in[3];
  const float* c0      = (const float*)d_in[4];
  const float* w_read  = (const float*)d_in[5];
  const float* w_write = (const float*)d_in[6];
  const float* W_ih    = (const float*)d_in[7];
  const float* W_hh    = (const float*)d_in[8];
  const float* b_ih    = (const float*)d_in[9];
  const float* b_hh    = (const float*)d_in[10];
  const float* W_rh    = (const float*)d_in[11];
  const float* b_rh    = (const float*)d_in[12];
  const float* W_wh    = (const float*)d_in[13];
  const float* b_wh    = (const float*)d_in[14];
  const float* W_o     = (const float*)d_in[15];
  const float* b_o     = (const float*)d_in[16];
  const int*   ind     = (const int*)d_in[17];

  float* out_  = (float*)d_out;              // tuple, concatenated flat (f32)
  float* o_out = out_ + 0;                   // (B,256)
  float* o_res = out_ + 65536;               // (B,64)
  float* o_om  = out_ + 81920;               // (B,64,2048)
  float* o_r   = out_ + 33636352;            // (B,64)
  float* o_h1  = out_ + 33652736;            // (B,512)
  float* o_c1  = out_ + 33783808;            // (B,512)
  float* o_wr  = out_ + 33914880;            // (B,2048)
  float* o_ww  = out_ + 34439168;            // (B,2048)

  float* ws      = (float*)d_ws;
  float* w_inp   = ws;                       // B*320
  float* w_gates = ws + 81920;               // B*2048
  float* w_hcat  = w_gates + 524288;         // B*576  ([h1|r])
  float* w_pr    = w_hcat + 147456;          // B*70
  float* w_pw    = w_pr + 17920;             // B*198

  k_concat    <<<(B_ * DIN_ + 255) / 256, 256, 0, stream>>>(x, prev_r, w_inp);
  k_gates_wmma<<<512, 128, 0, stream>>>(w_inp, h0, W_ih, W_hh, b_ih, b_hh, w_gates);
  k_lstm      <<<(B_ * CTRL_ + 255) / 256, 256, 0, stream>>>(w_gates, c0, o_h1, o_c1, w_hcat);
  k_heads     <<<(B_ * (PRW_ + PWW_) + 255) / 256, 256, 0, stream>>>(w_hcat, W_rh, b_rh,
                                                                     W_wh, b_wh, w_pr, w_pw);
  k_address   <<<B_, 256, 0, stream>>>(mem, w_pr, w_pw, w_read, w_write,
                                       o_wr, o_ww, o_r, w_hcat);
  k_out_wmma  <<<64, 128, 0, stream>>>(w_hcat, W_o, b_o, o_out);
  k_update_mem<<<B_ * 32, 256, 0, stream>>>(mem, w_pw, o_ww, o_om);
  k_residual  <<<B_, 256, 0, stream>>>(ind, o_om, o_res);
}